// ScaledDotproductAttention_14113262535512
// MI455X (gfx1250) — compile-verified
//
#include <hip/hip_runtime.h>

// ---------------------------------------------------------------------------
// Causal attention with relative-position bias, B=1 H=16 S=2048 DK=DV=64 (f32).
// Returns BOTH output [1,16,2048,64] and attn [1,16,2048,2048], so the full
// probability matrix is materialized: bound by 256MB bias reads + 256MB attn
// writes (~22us floor @ 23.3 TB/s HBM).
//
// CDNA5 paths used:
//   * V_WMMA_F32_16X16X4_F32 for both GEMMs (native f32 matrix op, wave32)
//   * GLOBAL_LOAD_ASYNC_TO_LDS_B128 to DMA the bias tile into LDS, issued
//     before the QK^T loop so the dominant HBM read overlaps WMMA compute
//     (ASYNCcnt / s_wait_asynccnt)
//   * non-temporal b128 stores for the attn stream (don't thrash 192MB L2,
//     which serves K/V re-reads across the 128 q-tiles per head)
//
// One workgroup (8 wave32) per (head, 16-query tile); the 16x2048 score block
// and the 16x2048 bias tile both live in LDS (~271KB of the WGP's 320KB).
// ---------------------------------------------------------------------------

typedef float v2f __attribute__((ext_vector_type(2)));
typedef float v4f __attribute__((ext_vector_type(4)));
typedef float v8f __attribute__((ext_vector_type(8)));

#define HH       16
#define SS       2048
#define DKK      64
#define DVV      64
#define QTILE    16
// Padded LDS row strides (floats). SSTRIDE%64==4, QSTRIDE%64==4 so the 16
// lanes of a half-wave reading one column-pair of 16 rows hit distinct bank
// groups -> conflict-free ds_load_b64 fragment reads.
#define SSTRIDE  2052
#define QSTRIDE  68

static constexpr int SMEM_FLOATS =
    16 * QSTRIDE      // staged Q tile
  + 16 * SSTRIDE      // score / probability row-block
  + 16 * SS           // async-DMA'd bias tile
  + 32                // rowmax[16] + rowinv[16]
  + 4 * 256;          // P*V partial accumulators

__global__ __launch_bounds__(256)
void attn_fused_wmma_f32(const float* __restrict__ qg,
                         const float* __restrict__ kg,
                         const float* __restrict__ vg,
                         const float* __restrict__ biasg,
                         float* __restrict__ outg,
                         float* __restrict__ attng)
{
    extern __shared__ float smem[];
    float* qs      = smem;                   // [16][QSTRIDE]
    float* sc      = qs + 16 * QSTRIDE;      // [16][SSTRIDE]
    float* biasbuf = sc + 16 * SSTRIDE;      // [16][SS]
    float* rowmax  = biasbuf + 16 * SS;      // [16]
    float* rowinv  = rowmax + 16;            // [16]
    float* part    = rowinv + 16;            // [4][256]

    const int h    = blockIdx.x >> 7;        // head
    const int qt   = blockIdx.x & 127;       // query tile index
    const int q0   = qt << 4;
    const int tid  = threadIdx.x;
    const int w    = tid >> 5;               // wave id 0..7
    const int lane = tid & 31;
    const int half = lane >> 4;              // WMMA fragment K-select
    const int l16  = lane & 15;              // M/N index within fragment

    const float scale = 0.125f;              // 1/sqrt(64)

    // ---- Phase A: kick off async DMA of the 16x2048 bias tile into LDS ----
    // 32 global_load_async_to_lds_b128 per wave (ASYNCcnt <= 32 < 63 cap);
    // the copy streams from HBM while the waves run the QK^T WMMA loop below.
    {
        const float* bsrc = biasg + (size_t)(h * SS + q0) * SS;
        for (int i = 0; i < 32; ++i) {
            const int f4  = (i << 8) + tid;        // float4 index, 0..8191
            const int row = f4 >> 9;               // 512 float4s per row
            const int col = (f4 & 511) << 2;
            const unsigned int  ldst = (unsigned int)(uintptr_t)(biasbuf + row * SS + col);
            const unsigned long long gsrc =
                (unsigned long long)(uintptr_t)(bsrc + (size_t)row * SS + col);
            asm volatile("global_load_async_to_lds_b128 %0, %1, off"
                         :: "v"(ldst), "v"(gsrc) : "memory");
        }
    }

    // ---- Phase 0: stage the 16x64 Q tile into LDS (coalesced b128) ----
    {
        const int row = tid >> 4;            // 0..15
        const int c   = (tid & 15) << 2;     // 0,4,...,60
        v4f x = *(const v4f*)(qg + ((size_t)(h * SS + q0 + row) * DKK + c));
        *(v4f*)(qs + row * QSTRIDE + c) = x;
    }
    __syncthreads();

    // Preload all A (Q) fragments into registers; reused for every key tile.
    // ISA 32-bit A 16x4 layout: lane l16 = row M; lanes<16 hold K={0,1},
    // lanes>=16 hold K={2,3}.
    v2f afrag[16];
#pragma unroll
    for (int c = 0; c < 16; ++c)
        afrag[c] = *(const v2f*)(qs + l16 * QSTRIDE + 4 * c + 2 * half);

    // ---- Phase 1: raw scores = scale * Q K^T into LDS (causal tiles only) --
    for (int kt = w; kt <= qt; kt += 8) {
        const int k0 = kt << 4;
        // B (K^T) fragment: b[j] = K[key = k0+l16][4c + 2*half + j]
        const float* krow = kg + ((size_t)(h * SS + k0 + l16) * DKK + 2 * half);
        v8f acc = {};
#pragma unroll
        for (int c = 0; c < 16; ++c) {
            v2f b = *(const v2f*)(krow + 4 * c);
            acc = __builtin_amdgcn_wmma_f32_16x16x4_f32(
                false, afrag[c], false, b, (short)0, acc, false, false);
        }
        // C/D layout: VGPR r -> row M = r + 8*half, col N = l16.
#pragma unroll
        for (int r = 0; r < 8; ++r) {
            const int m = r + 8 * half;
            sc[m * SSTRIDE + k0 + l16] = acc[r] * scale;
        }
    }

    // Bias DMA must have landed before the add pass; then sync all waves.
    asm volatile("s_wait_asynccnt 0x0" ::: "memory");
    __syncthreads();

    const int L = (qt + 1) << 4;             // valid key extent for this tile

    // ---- Phase 1.5: scores += bias (from LDS), apply causal mask ----------
    // Mask AFTER bias so masked entries are exactly -1e9 (reference order).
    for (int row = 0; row < 16; ++row) {
        const int qrow = q0 + row;
        float* r       = sc + row * SSTRIDE;
        const float* b = biasbuf + row * SS;
        const int nc4  = L >> 2;
        for (int c4 = tid; c4 < nc4; c4 += 256) {
            const int c = c4 << 2;
            v4f s = *(const v4f*)(r + c);
            v4f bb = *(const v4f*)(b + c);
            s += bb;
#pragma unroll
            for (int j = 0; j < 4; ++j)
                if (c + j > qrow) s[j] = -1e9f;
            *(v4f*)(r + c) = s;
        }
    }
    __syncthreads();

    // ---- Phase 2: per-row max and 1/sum (one wave per row, shfl reduce) ----
    for (int row = w; row < 16; row += 8) {
        const float* r = sc + row * SSTRIDE;
        float m = -3.0e38f;
        for (int k = lane; k < L; k += 32) m = fmaxf(m, r[k]);
#pragma unroll
        for (int off = 16; off; off >>= 1) m = fmaxf(m, __shfl_xor(m, off, 32));
        float s = 0.0f;
        for (int k = lane; k < L; k += 32) s += expf(r[k] - m);
#pragma unroll
        for (int off = 16; off; off >>= 1) s += __shfl_xor(s, off, 32);
        if (lane == 0) { rowmax[row] = m; rowinv[row] = 1.0f / s; }
    }
    __syncthreads();

    // ---- Phase 3: normalize; stream attn to HBM (non-temporal b128) -------
    // Masked region (c >= L) written as exact zeros (expf(-1e9-max) == 0),
    // matching the reference softmax bit-for-bit.
    float* attnb = attng + (size_t)(h * SS + q0) * SS;
    for (int row = 0; row < 16; ++row) {
        const float m   = rowmax[row];
        const float inv = rowinv[row];
        float* r    = sc + row * SSTRIDE;
        float* gout = attnb + (size_t)row * SS;
        for (int c4 = tid; c4 < SS / 4; c4 += 256) {
            const int c = c4 << 2;
            v4f p = {};
            if (c < L) {                     // L is a multiple of 16
                v4f sv = *(const v4f*)(r + c);
#pragma unroll
                for (int j = 0; j < 4; ++j) p[j] = expf(sv[j] - m) * inv;
                *(v4f*)(r + c) = p;          // P reused by phase 4
            }
            __builtin_nontemporal_store(p, (v4f*)(gout + c));
        }
    }
    __syncthreads();

    // ---- Phase 4: out = P x V via WMMA. 8 waves = 4 col-blocks x 2 halves --
    const int cb  = w & 3;                   // output column block (16 cols)
    const int hf  = w >> 2;                  // K-range half
    const int nc2 = L >> 3;                  // chunks of 4 per half
    v8f acc = {};
    for (int cc = hf * nc2; cc < hf * nc2 + nc2; ++cc) {
        const int kc = cc << 2;
        // A = P fragment from LDS (row l16, K-pair selected by lane half)
        v2f a = *(const v2f*)(sc + l16 * SSTRIDE + kc + 2 * half);
        // B = V fragment: b[j] = V[kc + 2*half + j][16*cb + l16]
        const float* vrow = vg + (size_t)(h * SS + kc + 2 * half) * DVV + cb * 16 + l16;
        v2f b; b[0] = vrow[0]; b[1] = vrow[DVV];
        acc = __builtin_amdgcn_wmma_f32_16x16x4_f32(
            false, a, false, b, (short)0, acc, false, false);
    }
    if (hf == 1) {
#pragma unroll
        for (int r = 0; r < 8; ++r) part[cb * 256 + r * 32 + lane] = acc[r];
    }
    __syncthreads();
    if (hf == 0) {
        float* ob = outg + (size_t)(h * SS + q0) * DVV + cb * 16 + l16;
#pragma unroll
        for (int r = 0; r < 8; ++r) {
            const int m = r + 8 * half;
            ob[(size_t)m * DVV] = acc[r] + part[cb * 256 + r * 32 + lane];
        }
    }
}

extern "C" void kernel_launch(void* const* d_in, const int* in_sizes, int n_in,
                              void* d_out, int out_size, void* d_ws, size_t ws_size,
                              hipStream_t stream) {
    (void)in_sizes; (void)n_in; (void)d_ws; (void)ws_size; (void)out_size;
    const float* q    = (const float*)d_in[0];
    const float* k    = (const float*)d_in[1];
    const float* v    = (const float*)d_in[2];
    // d_in[3] is the causal (tril) bool mask: fixed structure, applied
    // analytically (k > q -> -1e9).
    const float* bias = (const float*)d_in[4];

    float* out  = (float*)d_out;                       // [1,16,2048,64]
    float* attn = out + (size_t)HH * SS * DVV;         // [1,16,2048,2048]

    const dim3 grid(HH * (SS / QTILE));                // 2048 workgroups
    const dim3 block(256);                             // 8 wave32
    const size_t smem = (size_t)SMEM_FLOATS * sizeof(float);  // ~271 KB

    hipLaunchKernelGGL(attn_fused_wmma_f32, grid, block, smem, stream,
                       q, k, v, bias, out, attn);
}